// SegDopplerLoss_20109036880107
// MI455X (gfx1250) — compile-verified
//
#include <hip/hip_runtime.h>
#include <stdint.h>

// Problem dims (from reference)
#define BS   2
#define DOP  32
#define RAN  128
#define AZI  128
#define ELE  32
#define NVOX (RAN*AZI*ELE)        // 524288 voxels per batch
#define VCH  128                  // voxels per chunk (TDM tile width)
#define TPB  128                  // threads per block = 1 chunk
#define CHPB (NVOX / VCH)         // 4096 chunks per batch
#define NCH  (BS * CHPB)          // 8192 chunks total
#define GRID 1024                 // blocks; each handles NCH/GRID = 8 chunks

typedef unsigned int u32;
typedef unsigned long long u64;
typedef u32 u32x4 __attribute__((ext_vector_type(4)));
typedef int i32x4 __attribute__((ext_vector_type(4)));
typedef int i32x8 __attribute__((ext_vector_type(8)));

__device__ __forceinline__ float fast_sigmoid(float x) {
  return 1.0f / (1.0f + __expf(-x));
}

// ---------------------------------------------------------------------------
// Kernel 1: spherical -> cartesian precompute into workspace; zero accumulator
// ---------------------------------------------------------------------------
__global__ void xyz_kernel(const float* __restrict__ coords,
                           float* __restrict__ xyz,
                           float* __restrict__ accum)
{
  int idx = blockIdx.x * blockDim.x + threadIdx.x;
  if (idx == 0) accum[0] = 0.0f;
  if (idx >= BS * NVOX) return;
  int b = idx / NVOX, v = idx - b * NVOX;
  size_t base = (size_t)b * 3 * NVOX + v;
  float r = coords[base];
  float a = coords[base + NVOX];
  float e = coords[base + 2 * NVOX];
  float ce = __cosf(e);
  xyz[base]            = r * ce * __cosf(a);
  xyz[base + NVOX]     = r * ce * __sinf(a);
  xyz[base + 2 * NVOX] = r * __sinf(e);
}

// ---------------------------------------------------------------------------
// TDM: async 2D tile load (DOP rows x VCH floats, row stride NVOX) into LDS.
// Descriptor packed per CDNA5 ISA.
// ---------------------------------------------------------------------------
__device__ __forceinline__ void tdm_load_x1_tile(const float* gaddr, u32 lds_byte)
{
  u64 ga = (u64)(uintptr_t)gaddr;
  u32x4 g0;
  g0[0] = 1u;                                              // count=1, user D#
  g0[1] = lds_byte;                                        // lds_addr (bytes)
  g0[2] = (u32)ga;                                         // global_addr[31:0]
  g0[3] = ((u32)(ga >> 32) & 0x01FFFFFFu) | 0x80000000u;   // addr[56:32] | type=2
  i32x8 g1;
  g1[0] = (int)(2u << 16);                                 // data_size=4B; mask=0
  g1[1] = (int)((u32)VCH << 16);                           // tensor_dim0[15:0]
  g1[2] = (int)((u32)DOP << 16);                           // dim0 hi=0 | tensor_dim1 lo
  g1[3] = (int)((u32)VCH << 16);                           // dim1 hi=0 | tile_dim0
  g1[4] = (int)DOP;                                        // tile_dim1; tile_dim2=0
  g1[5] = (int)(u32)NVOX;                                  // tensor_dim0_stride lo32
  g1[6] = 0;                                               // stride0 hi | stride1 lo
  g1[7] = 0;
  i32x4 z4 = {0, 0, 0, 0};
#if defined(__clang_major__) && (__clang_major__ >= 23)
  i32x8 z8 = {0, 0, 0, 0, 0, 0, 0, 0};
  __builtin_amdgcn_tensor_load_to_lds(g0, g1, z4, z4, z8, 0);
#else
  __builtin_amdgcn_tensor_load_to_lds(g0, g1, z4, z4, 0);
#endif
}

// ---------------------------------------------------------------------------
// Kernel 2: fused main kernel. TDM double-buffers x1 tiles into LDS while the
// waves do the per-voxel softmax/argmax, trilinear gather, and loss terms.
// ---------------------------------------------------------------------------
__global__ void __launch_bounds__(TPB)
main_kernel(const float* __restrict__ seg,
            const float* __restrict__ flow,
            const float* __restrict__ x1,
            const float* __restrict__ dop_arr,
            const float* __restrict__ ori,
            const float* __restrict__ gumbels,
            const float* __restrict__ xyz,
            float* __restrict__ accum)
{
  __shared__ float bufX[2][DOP * VCH];   // 2 x 16KB double-buffered x1 tiles
  __shared__ float sdop[BS * DOP];
  __shared__ float red[TPB];

  const int t = threadIdx.x;
  if (t < BS * DOP) sdop[t] = dop_arr[t];

  const bool issuer = (t < 32);          // wave 0 drives the TDM
  const u32 ldsX0 = (u32)(uintptr_t)&bufX[0][0];
  const u32 ldsX1 = (u32)(uintptr_t)&bufX[1][0];

  float local = 0.0f;

  int chunk = blockIdx.x;
  if (issuer) {                          // prologue: fetch first tile
    int b = chunk / CHPB, v0 = (chunk - b * CHPB) * VCH;
    tdm_load_x1_tile(x1 + (size_t)b * DOP * NVOX + v0, ldsX0);
  }
  int buf = 0;
  for (; chunk < NCH; chunk += GRID) {
    const int nchunk = chunk + GRID;
    if (issuer) {
      if (nchunk < NCH) {                // prefetch next tile, wait for current
        int nb = nchunk / CHPB, nv0 = (nchunk - nb * CHPB) * VCH;
        tdm_load_x1_tile(x1 + (size_t)nb * DOP * NVOX + nv0, buf ? ldsX0 : ldsX1);
        __builtin_amdgcn_s_wait_tensorcnt(1);
      } else {
        __builtin_amdgcn_s_wait_tensorcnt(0);
      }
    }
    __syncthreads();                     // tile visible to all 4 waves

    const int b  = chunk / CHPB;
    const int v0 = (chunk - b * CHPB) * VCH;
    const int v  = v0 + t;
    const float* __restrict__ bX = &bufX[buf][0];
    const float* __restrict__ gP = gumbels + (size_t)b * DOP * NVOX + v0;

    // ---- pass 1: max(x1) and argmax(x1+gumbels) (gumbels streamed once) ----
    float mx = -3.0e38f, m1 = -3.0e38f;
    int am = 0;
#pragma unroll
    for (int d = 0; d < DOP; ++d) {
      float xv = bX[d * VCH + t];
      float gv = gP[(size_t)d * NVOX + t];
      float s  = xv + gv;
      if (s > m1) { m1 = s; am = d; }
      mx = fmaxf(mx, xv);
    }
    // ---- pass 2: softmax(x1) dotted with dop_arr ----
    float se = 0.0f, swe = 0.0f;
#pragma unroll
    for (int d = 0; d < DOP; ++d) {
      float e = __expf(bX[d * VCH + t] - mx);
      se  += e;
      swe += e * sdop[b * DOP + d];
    }
    const float x1dop1 = sdop[b * DOP + am];
    const float x1dop2 = swe / se;

    // ---- warp + trilinear sample of precomputed xyz (border clamp) ----
    const int e_i = v & (ELE - 1);
    const int a_i = (v >> 5) & (AZI - 1);
    const int r_i = v >> 12;
    const size_t fb = ((size_t)b * NVOX + v) * 3;
    const float ix = fminf(fmaxf((float)e_i + flow[fb + 0], 0.f), (float)(ELE - 1));
    const float iy = fminf(fmaxf((float)a_i + flow[fb + 1], 0.f), (float)(AZI - 1));
    const float iz = fminf(fmaxf((float)r_i + flow[fb + 2], 0.f), (float)(RAN - 1));
    const float x0f = floorf(ix), y0f = floorf(iy), z0f = floorf(iz);
    const float wx = ix - x0f, wy = iy - y0f, wz = iz - z0f;
    const int x0 = (int)x0f, y0 = (int)y0f, z0 = (int)z0f;
    const int xx2[2] = { x0, min(x0 + 1, ELE - 1) };
    const int yy2[2] = { y0, min(y0 + 1, AZI - 1) };
    const int zz2[2] = { z0, min(z0 + 1, RAN - 1) };
    const float wxv[2] = { 1.f - wx, wx };
    const float wyv[2] = { 1.f - wy, wy };
    const float wzv[2] = { 1.f - wz, wz };

    const float* __restrict__ X = xyz + (size_t)b * 3 * NVOX;
    float sx = 0.f, sy = 0.f, sz = 0.f;
#pragma unroll
    for (int kz = 0; kz < 2; ++kz)
#pragma unroll
      for (int ky = 0; ky < 2; ++ky)
#pragma unroll
        for (int kx = 0; kx < 2; ++kx) {
          const float wk = wzv[kz] * wyv[ky] * wxv[kx];
          const int id = (zz2[kz] * AZI + yy2[ky]) * ELE + xx2[kx];
          sx += wk * X[id];
          sy += wk * X[id + NVOX];
          sz += wk * X[id + 2 * NVOX];
        }
    const float ox = X[v], oy = X[v + NVOX], oz = X[v + 2 * NVOX];

    // ---- velocity, doppler label, loss term ----
    const size_t ob = (size_t)b * 3 * NVOX + v;
    const float dlabel = (sx - ox) * 10.f * ori[ob]
                       + (sy - oy) * 10.f * ori[ob + NVOX]
                       + (sz - oz) * 10.f * ori[ob + 2 * NVOX];
    float e1 = dlabel - x1dop1; e1 *= e1;
    float e2 = dlabel - x1dop2; e2 *= e2;
    const float slab = 0.5f * (fast_sigmoid(1.5f - 10.f * e1) +
                               fast_sigmoid(1.5f - 10.f * e2));
    const float ssig = fast_sigmoid(seg[(size_t)b * NVOX + v]);
    local += fabsf(ssig - slab);

    __syncthreads();                     // all reads of bufX[buf] retired
    buf ^= 1;                            // before next TDM overwrites it
  }

  // ---- block reduction + global accumulate ----
  red[t] = local;
  __syncthreads();
#pragma unroll
  for (int s = TPB / 2; s > 0; s >>= 1) {
    if (t < s) red[t] += red[t + s];
    __syncthreads();
  }
  if (t == 0) atomicAdd(accum, red[0]);
}

// ---------------------------------------------------------------------------
// Kernel 3: final mean
// ---------------------------------------------------------------------------
__global__ void finalize_kernel(const float* __restrict__ accum,
                                float* __restrict__ out)
{
  if (threadIdx.x == 0 && blockIdx.x == 0)
    out[0] = accum[0] * (1.0f / (float)(BS * NVOX));
}

// ---------------------------------------------------------------------------
extern "C" void kernel_launch(void* const* d_in, const int* in_sizes, int n_in,
                              void* d_out, int out_size, void* d_ws, size_t ws_size,
                              hipStream_t stream)
{
  (void)in_sizes; (void)n_in; (void)out_size; (void)ws_size;
  const float* seg    = (const float*)d_in[0];
  const float* flow   = (const float*)d_in[1];
  const float* x1     = (const float*)d_in[2];
  const float* dopA   = (const float*)d_in[3];
  const float* coords = (const float*)d_in[4];
  const float* ori    = (const float*)d_in[5];
  const float* gumb   = (const float*)d_in[6];

  float* xyz   = (float*)d_ws;                       // 2*3*524288 floats (~12.6MB)
  float* accum = xyz + (size_t)BS * 3 * NVOX;        // +1 float
  float* out   = (float*)d_out;

  const int total = BS * NVOX;
  xyz_kernel<<<(total + 255) / 256, 256, 0, stream>>>(coords, xyz, accum);
  main_kernel<<<GRID, TPB, 0, stream>>>(seg, flow, x1, dopA, ori, gumb, xyz, accum);
  finalize_kernel<<<1, 32, 0, stream>>>(accum, out);
}